// CTSFNOBlock_34677565948285
// MI455X (gfx1250) — compile-verified
//
#include <hip/hip_runtime.h>
#include <math.h>

typedef __attribute__((ext_vector_type(2))) float v2f;
typedef __attribute__((ext_vector_type(8))) float v8f;

#define PI_D 3.14159265358979323846

// ---------------- workspace layout (units: floats) ----------------
constexpr long long SZ_A1    = 255LL*128*128;  // Pw(L=128), layout [mfull][l][t]
constexpr long long SZ_XP    = 128LL*256*128;  // x zero-padded  [t][p(256)][c]
constexpr long long SZ_PH1   = 256LL*256;      // phase1 tables [m(256)][p(256)]
constexpr long long SZ_G     = 128LL*256*128;  // g  [t][m(256)][c]
constexpr long long SZ_F1    = 255LL*128*128;  // flm1 [m][l][c]
constexpr long long SZ_R1    = 64LL*128;
constexpr long long SZ_R2P   = 128LL*256;
constexpr long long SZ_T1    = 256LL*64*128;   // [m(256)][i][c]
constexpr long long SZ_FF    = 64LL*128*128;   // flm resized [l][m(128)][c]
constexpr long long SZ_WC    = 64LL*128*128;   // t_c * spec_w  [l][i][o]
constexpr long long SZ_F2    = 64LL*128*128;   // mixed flm [l][m(128)][o]
constexpr long long SZ_P2    = 127LL*64*64;
constexpr long long SZ_FT    = 127LL*64*128;   // f_t [m][t][c]
constexpr long long SZ_PH2   = 128LL*128;
constexpr long long SZ_XSPEC = 64LL*128*128;   // [t][p(128)][c]
constexpr long long SZ_TVEC  = 256;
constexpr long long SZ_TC    = 64;
constexpr long long SZ_WSP   = 128LL*128;
constexpr long long SZ_XS1   = 128LL*255*128;  // [t][p][c]
constexpr long long SZ_T2    = 256LL*64*128;   // [p(256)][i][c]
constexpr long long SZ_XS2   = 64LL*128*128;   // [i][j(128)][c]
constexpr long long SZ_XSP   = 67LL*129*128;   // conv zero-padded image
constexpr long long SZ_XCV   = 64LL*128*128;   // conv out [l][m(128)][o]

constexpr long long OFF_A1    = 0;
constexpr long long OFF_XP    = OFF_A1    + SZ_A1;
constexpr long long OFF_COS1  = OFF_XP    + SZ_XP;
constexpr long long OFF_SIN1N = OFF_COS1  + SZ_PH1;
constexpr long long OFF_GR    = OFF_SIN1N + SZ_PH1;
constexpr long long OFF_GI    = OFF_GR    + SZ_G;
constexpr long long OFF_F1R   = OFF_GI    + SZ_G;
constexpr long long OFF_F1I   = OFF_F1R   + SZ_F1;
constexpr long long OFF_R1    = OFF_F1I   + SZ_F1;
constexpr long long OFF_R2P   = OFF_R1    + SZ_R1;
constexpr long long OFF_T1R   = OFF_R2P   + SZ_R2P;
constexpr long long OFF_T1I   = OFF_T1R   + SZ_T1;
constexpr long long OFF_FFR   = OFF_T1I   + SZ_T1;
constexpr long long OFF_FFI   = OFF_FFR   + SZ_FF;
constexpr long long OFF_WCR   = OFF_FFI   + SZ_FF;
constexpr long long OFF_WCI   = OFF_WCR   + SZ_WC;
constexpr long long OFF_F2R   = OFF_WCI   + SZ_WC;
constexpr long long OFF_F2I   = OFF_F2R   + SZ_F2;
constexpr long long OFF_P2A   = OFF_F2I   + SZ_F2;
constexpr long long OFF_P2B   = OFF_P2A   + SZ_P2;
constexpr long long OFF_FTR   = OFF_P2B   + SZ_P2;
constexpr long long OFF_FTI   = OFF_FTR   + SZ_FT;   // stage6 k=127 over-read of FTR lands here (finite, x0)
constexpr long long OFF_COS2  = OFF_FTI   + SZ_FT;
constexpr long long OFF_SIN2  = OFF_COS2  + SZ_PH2;
constexpr long long OFF_XSPEC = OFF_SIN2  + SZ_PH2;
constexpr long long OFF_TVEC  = OFF_XSPEC + SZ_XSPEC;
constexpr long long OFF_TCR   = OFF_TVEC  + SZ_TVEC;
constexpr long long OFF_TCI   = OFF_TCR   + SZ_TC;
constexpr long long OFF_WSP   = OFF_TCI   + SZ_TC;
constexpr long long OFF_XS1   = OFF_WSP   + SZ_WSP;
constexpr long long OFF_T2R   = OFF_XS1   + SZ_XS1;
constexpr long long OFF_XS2   = OFF_T2R   + SZ_T2;
constexpr long long OFF_XSP   = OFF_XS2   + SZ_XS2;
constexpr long long OFF_XCV   = OFF_XSP   + SZ_XSP;

// ---------------- generic strided-batched WMMA GEMM (N == 128) ----------------
// C[b][i][j] (=/+)= alpha * sum_k A[b][i][k] * B[b][k][j]
// Wave computes a 16(M) x 64(N) tile via 4 f32 16x16x4 accumulators.
__global__ void gemm_wmma(const float* __restrict__ A, const float* __restrict__ B,
                          float* __restrict__ C, int M, int K,
                          int lda, int ldb, int ldc,
                          long long sAb, long long sBb, long long sCb,
                          float alpha, int accumulate) {
  const int lane = threadIdx.x;            // wave32, blockDim = 32
  const int n0 = blockIdx.x * 64;
  const int m0 = blockIdx.y * 16;
  const long long b = blockIdx.z;
  A += b * sAb; B += b * sBb; C += b * sCb;
  const int half = lane >> 4;              // 0: lanes 0-15, 1: lanes 16-31
  const int l16  = lane & 15;
  v8f acc0 = {}, acc1 = {}, acc2 = {}, acc3 = {};
  const float* arow = A + (long long)(m0 + l16) * lda + 2 * half;   // K = 2*half, 2*half+1
  const float* brow = B + (long long)(2 * half) * ldb + n0 + l16;
  for (int k0 = 0; k0 < K; k0 += 4) {
    v2f a; a.x = arow[k0]; a.y = arow[k0 + 1];
    const float* bp = brow + (long long)k0 * ldb;
    v2f b0; b0.x = bp[0];   b0.y = bp[ldb];
    v2f b1; b1.x = bp[16];  b1.y = bp[ldb + 16];
    v2f b2; b2.x = bp[32];  b2.y = bp[ldb + 32];
    v2f b3; b3.x = bp[48];  b3.y = bp[ldb + 48];
    acc0 = __builtin_amdgcn_wmma_f32_16x16x4_f32(false, a, false, b0, (short)0, acc0, false, false);
    acc1 = __builtin_amdgcn_wmma_f32_16x16x4_f32(false, a, false, b1, (short)0, acc1, false, false);
    acc2 = __builtin_amdgcn_wmma_f32_16x16x4_f32(false, a, false, b2, (short)0, acc2, false, false);
    acc3 = __builtin_amdgcn_wmma_f32_16x16x4_f32(false, a, false, b3, (short)0, acc3, false, false);
    if (k0 + 4 < K) {
      __builtin_prefetch(arow + k0 + 4, 0, 1);
      __builtin_prefetch(bp + 4LL * ldb, 0, 1);
    }
  }
#pragma unroll
  for (int v = 0; v < 8; ++v) {
    const int row = m0 + half * 8 + v;     // C layout: VGPR v -> M = v + 8*half
    float* cp = C + (long long)row * ldc + n0 + l16;
    if (accumulate) {
      cp[0]  += alpha * acc0[v]; cp[16] += alpha * acc1[v];
      cp[32] += alpha * acc2[v]; cp[48] += alpha * acc3[v];
    } else {
      cp[0]  = alpha * acc0[v]; cp[16] = alpha * acc1[v];
      cp[32] = alpha * acc2[v]; cp[48] = alpha * acc3[v];
    }
  }
}

// ---------------- 3x3 SAME conv as 9 accumulated WMMA GEMMs ----------------
// xsp: zero-padded (67 x 129 x 128); ck: (3,3,128,128) HWIO; out: [l][m(128)][o]
__global__ void conv3x3_wmma(const float* __restrict__ xsp, const float* __restrict__ ck,
                             const float* __restrict__ cb, const float* __restrict__ tvec,
                             float* __restrict__ out) {
  const int lane = threadIdx.x;
  const int n0 = blockIdx.x * 64;
  const int m0 = blockIdx.y * 16;
  const int l  = blockIdx.z;
  const int half = lane >> 4, l16 = lane & 15;
  v8f acc0 = {}, acc1 = {}, acc2 = {}, acc3 = {};
  for (int dy = 0; dy < 3; ++dy) {
    for (int dx = 0; dx < 3; ++dx) {
      const float* arow = xsp + (((long long)(l + dy)) * 129 + (m0 + l16 + dx)) * 128 + 2 * half;
      const float* brow = ck + (((long long)(dy * 3 + dx)) * 128 + 2 * half) * 128 + n0 + l16;
      for (int k0 = 0; k0 < 128; k0 += 4) {
        v2f a; a.x = arow[k0]; a.y = arow[k0 + 1];
        const float* bp = brow + (long long)k0 * 128;
        v2f b0; b0.x = bp[0];  b0.y = bp[128];
        v2f b1; b1.x = bp[16]; b1.y = bp[144];
        v2f b2; b2.x = bp[32]; b2.y = bp[160];
        v2f b3; b3.x = bp[48]; b3.y = bp[176];
        acc0 = __builtin_amdgcn_wmma_f32_16x16x4_f32(false, a, false, b0, (short)0, acc0, false, false);
        acc1 = __builtin_amdgcn_wmma_f32_16x16x4_f32(false, a, false, b1, (short)0, acc1, false, false);
        acc2 = __builtin_amdgcn_wmma_f32_16x16x4_f32(false, a, false, b2, (short)0, acc2, false, false);
        acc3 = __builtin_amdgcn_wmma_f32_16x16x4_f32(false, a, false, b3, (short)0, acc3, false, false);
      }
    }
  }
  const int c0 = n0 + l16;
  const float bias0 = cb[c0]      + tvec[128 + c0];
  const float bias1 = cb[c0 + 16] + tvec[128 + c0 + 16];
  const float bias2 = cb[c0 + 32] + tvec[128 + c0 + 32];
  const float bias3 = cb[c0 + 48] + tvec[128 + c0 + 48];
#pragma unroll
  for (int v = 0; v < 8; ++v) {
    const int row = m0 + half * 8 + v;
    float* op = out + (((long long)l) * 128 + row) * 128 + c0;
    op[0]  = acc0[v] + bias0; op[16] = acc1[v] + bias1;
    op[32] = acc2[v] + bias2; op[48] = acc3[v] + bias3;
  }
}

// ---------------- small helper kernels ----------------
__global__ void zerof(float* p, long long n) {
  long long i = (long long)blockIdx.x * 256 + threadIdx.x;
  if (i < n) p[i] = 0.0f;
}

__global__ void build_xpad(const float* __restrict__ x, float* __restrict__ xp, long long n) {
  long long i = (long long)blockIdx.x * 256 + threadIdx.x;
  if (i >= n) return;
  long long t = i / 32768, r = i % 32768, p = r / 128, c = r % 128;
  xp[i] = (p < 255) ? x[(t * 255 + p) * 128 + c] : 0.0f;
}

__global__ void k_tvec(const float* __restrict__ temb,
                       const float* __restrict__ dw, const float* __restrict__ db,
                       const float* __restrict__ drw, const float* __restrict__ drb,
                       const float* __restrict__ diw, const float* __restrict__ dib,
                       float* tvec, float* tcr, float* tci) {
  int j = threadIdx.x;  // 256 threads
  float acc = db[j];
  for (int k = 0; k < 256; ++k) acc += temb[k] * dw[k * 256 + j];
  tvec[j] = acc;
  if (j < 64) {
    float ar = drb[j], ai = dib[j];
    for (int k = 0; k < 256; ++k) {
      ar += temb[k] * drw[k * 64 + j];
      ai += temb[k] * diw[k * 64 + j];
    }
    tcr[j] = ar; tci[j] = ai;
  }
}

__global__ void k_wc(const float* __restrict__ wr, const float* __restrict__ wi,
                     const float* __restrict__ tcr, const float* __restrict__ tci,
                     float* wcr, float* wci, long long n) {
  long long i = (long long)blockIdx.x * 256 + threadIdx.x;
  if (i >= n) return;
  int l = (int)(i / 16384);
  float tr = tcr[l], ti = tci[l], r = wr[i], im = wi[i];
  wcr[i] = tr * r - ti * im;
  wci[i] = tr * im + ti * r;
}

__global__ void k_wsp(const float* __restrict__ spw, const float* __restrict__ tvec,
                      float* wsp, long long n) {
  long long i = (long long)blockIdx.x * 256 + threadIdx.x;
  if (i >= n) return;
  int o = (int)(i % 128);
  wsp[i] = tvec[o] * spw[i];   // (w_t[None,:] * sp_weight)[i][o] = w_t[o]*sp_weight[i][o]
}

__global__ void k_phase1(float* c1, float* s1n, long long n) {
  long long i = (long long)blockIdx.x * 256 + threadIdx.x;
  if (i >= n) return;
  int m = (int)(i / 256), p = (int)(i % 256);
  if (m < 255 && p < 255) {
    double mval = (double)(m - 127);
    double phi = 2.0 * PI_D * (double)p / 255.0;
    double dphi = 2.0 * PI_D / 255.0;
    c1[i]  = (float)( cos(phi * mval) * dphi);   // Re(exp(-i phi m)) * dphi
    s1n[i] = (float)(-sin(phi * mval) * dphi);   // Im(exp(-i phi m)) * dphi
  } else { c1[i] = 0.0f; s1n[i] = 0.0f; }
}

__global__ void k_phase2(float* c2, float* s2, long long n) {
  long long i = (long long)blockIdx.x * 256 + threadIdx.x;
  if (i >= n) return;
  int p = (int)(i / 128), m = (int)(i % 128);
  if (p < 127 && m < 127) {
    double mval = (double)(m - 63);
    double phi = 2.0 * PI_D * (double)p / 127.0;
    c2[i] = (float)cos(mval * phi);
    s2[i] = (float)sin(mval * phi);
  } else { c2[i] = 0.0f; s2[i] = 0.0f; }
}

// antialiased bilinear resize weight rows (jax.image.resize, triangle kernel, normalized)
__global__ void k_resize(float* r1, float* r2p) {
  int tid = threadIdx.x;
  if (tid < 64) {                       // R1: 64 x 128, scale = 0.5
    int i = tid;
    float scale = 0.5f;
    float sample = (i + 0.5f) / scale - 0.5f;
    float sum = 0.0f;
    for (int j = 0; j < 128; ++j) {
      float w = 1.0f - fabsf(sample - j) * scale;
      if (w > 0.0f) sum += w;
    }
    for (int j = 0; j < 128; ++j) {
      float w = 1.0f - fabsf(sample - j) * scale;
      r1[i * 128 + j] = (w > 0.0f) ? (w / sum) : 0.0f;
    }
  } else if (tid < 192) {               // R2P: 128 x 256 (row 127 / col 255 are zero pad)
    int j = tid - 64;
    if (j < 127) {
      float scale = 127.0f / 255.0f;
      float sample = (j + 0.5f) / scale - 0.5f;
      float sum = 0.0f;
      for (int p = 0; p < 255; ++p) {
        float w = 1.0f - fabsf(sample - p) * scale;
        if (w > 0.0f) sum += w;
      }
      for (int p = 0; p < 255; ++p) {
        float w = 1.0f - fabsf(sample - p) * scale;
        r2p[j * 256 + p] = (w > 0.0f) ? (w / sum) : 0.0f;
      }
      r2p[j * 256 + 255] = 0.0f;
    } else {
      for (int p = 0; p < 256; ++p) r2p[j * 256 + p] = 0.0f;
    }
  }
}

// normalized associated Legendre, layout [mfull(2L-1)][l(L)][t(L)], m>=0 part.
// If weighted, seeds with quadrature weight w(t) = sin(theta)*pi/L (linear recurrence => scales all).
__global__ void k_legendre(float* out, int L, int weighted) {
  int t = threadIdx.x;
  if (t >= L) return;
  double theta = (2.0 * t + 1.0) * PI_D / (2.0 * L - 1.0);
  double x = cos(theta), s = sin(theta);
  double w = weighted ? (s * PI_D / (double)L) : 1.0;
  auto idx = [&](int m, int l) { return ((long long)(L - 1 + m) * L + l) * (long long)L + t; };
  out[idx(0, 0)] = (float)(sqrt(1.0 / (4.0 * PI_D)) * w);
  for (int l = 1; l < L; ++l) {
    double pprev = (double)out[idx(l - 1, l - 1)];
    out[idx(l, l)]     = (float)(-sqrt((2.0 * l + 1.0) / (2.0 * l)) * s * pprev);
    out[idx(l - 1, l)] = (float)( sqrt(2.0 * l + 1.0) * x * pprev);
    for (int m = 0; m <= l - 2; ++m) {
      double ll = (double)l, mm = (double)m;
      double a = sqrt((4.0 * ll * ll - 1.0) / (ll * ll - mm * mm));
      double b = sqrt((2.0 * ll + 1.0) * ((ll - 1.0) * (ll - 1.0) - mm * mm) /
                      ((2.0 * ll - 3.0) * (ll * ll - mm * mm)));
      out[idx(m, l)] = (float)(a * x * (double)out[idx(m, l - 1)] - b * (double)out[idx(m, l - 2)]);
    }
  }
}

__global__ void k_mirror(float* out, int L, long long n) {   // negative-m fill: (-1)^m mirror
  long long i = (long long)blockIdx.x * 256 + threadIdx.x;
  if (i >= n) return;
  int LL = L * L;
  int m = (int)(i / LL) + 1;
  int r = (int)(i % LL);
  int l = r / L, t = r % L;
  float v = out[((long long)(L - 1 + m) * L + l) * L + t];
  out[((long long)(L - 1 - m) * L + l) * L + t] = (m & 1) ? -v : v;
}

__global__ void k_transp_p2(const float* __restrict__ a, float* __restrict__ b, long long n) {
  long long i = (long long)blockIdx.x * 256 + threadIdx.x;   // b[(m*64+t)*64+l] = a[(m*64+l)*64+t]
  if (i >= n) return;
  int m = (int)(i / 4096), t = (int)((i / 64) % 64), l = (int)(i % 64);
  b[i] = a[((long long)m * 64 + l) * 64 + t];
}

__global__ void k_xsppad(const float* __restrict__ xs2, float* __restrict__ xsp, long long n) {
  long long i = (long long)blockIdx.x * 256 + threadIdx.x;
  if (i >= n) return;
  long long row = i / (129 * 128), r = i % (129 * 128);
  int col = (int)(r / 128), c = (int)(r % 128);
  int l = (int)row - 1, m = col - 1;
  xsp[i] = (l >= 0 && l < 64 && m >= 0 && m < 127) ? xs2[((long long)l * 128 + m) * 128 + c] : 0.0f;
}

// fused gelu(tanh) + channel LayerNorm; writes final (64,127,128) output
__global__ void k_fuse(const float* __restrict__ xspec, const float* __restrict__ xcv,
                       const float* __restrict__ lns, const float* __restrict__ lnb,
                       float* __restrict__ out) {
  __shared__ float sh[128];
  int c = threadIdx.x;
  int n = blockIdx.x;              // 64*127 positions
  int l = n / 127, m = n % 127;
  long long idx = ((long long)l * 128 + m) * 128 + c;
  float v = xspec[idx] + xcv[idx];
  float g = 0.5f * v * (1.0f + tanhf(0.7978845608028654f * (v + 0.044715f * v * v * v)));
  sh[c] = g; __syncthreads();
  for (int s = 64; s > 0; s >>= 1) { if (c < s) sh[c] += sh[c + s]; __syncthreads(); }
  float mu = sh[0] * (1.0f / 128.0f); __syncthreads();
  float d = g - mu;
  sh[c] = d * d; __syncthreads();
  for (int s = 64; s > 0; s >>= 1) { if (c < s) sh[c] += sh[c + s]; __syncthreads(); }
  float var = sh[0] * (1.0f / 128.0f);
  out[((long long)l * 127 + m) * 128 + c] = d * rsqrtf(var + 1e-6f) * lns[c] + lnb[c];
}

// ---------------- host side ----------------
static inline void launch_gemm(hipStream_t s, const float* A, const float* B, float* C,
                               int M, int K, int lda, int ldb, int ldc,
                               long long sA, long long sB, long long sC,
                               float alpha, int acc, int batch) {
  dim3 g(2, M / 16, batch);
  gemm_wmma<<<g, dim3(32), 0, s>>>(A, B, C, M, K, lda, ldb, ldc, sA, sB, sC, alpha, acc);
}

#define EWG(n) dim3((unsigned)(((n) + 255) / 256)), dim3(256), 0, stream

extern "C" void kernel_launch(void* const* d_in, const int* in_sizes, int n_in,
                              void* d_out, int out_size, void* d_ws, size_t ws_size,
                              hipStream_t stream) {
  (void)in_sizes; (void)n_in; (void)out_size; (void)ws_size;
  const float* x       = (const float*)d_in[0];
  const float* t_emb   = (const float*)d_in[1];
  const float* sp_dw   = (const float*)d_in[2];
  const float* sp_db   = (const float*)d_in[3];
  const float* sp_wt   = (const float*)d_in[4];
  const float* sp_ck   = (const float*)d_in[5];
  const float* sp_cb   = (const float*)d_in[6];
  const float* spec_wr = (const float*)d_in[7];
  const float* spec_wi = (const float*)d_in[8];
  const float* drw     = (const float*)d_in[9];
  const float* drb     = (const float*)d_in[10];
  const float* diw     = (const float*)d_in[11];
  const float* dib     = (const float*)d_in[12];
  const float* lns     = (const float*)d_in[13];
  const float* lnb     = (const float*)d_in[14];
  float* out = (float*)d_out;
  float* ws  = (float*)d_ws;

  float *A1 = ws + OFF_A1, *XP = ws + OFF_XP, *COS1 = ws + OFF_COS1, *SIN1N = ws + OFF_SIN1N;
  float *GR = ws + OFF_GR, *GI = ws + OFF_GI, *F1R = ws + OFF_F1R, *F1I = ws + OFF_F1I;
  float *R1 = ws + OFF_R1, *R2P = ws + OFF_R2P, *T1R = ws + OFF_T1R, *T1I = ws + OFF_T1I;
  float *FFR = ws + OFF_FFR, *FFI = ws + OFF_FFI, *WCR = ws + OFF_WCR, *WCI = ws + OFF_WCI;
  float *F2R = ws + OFF_F2R, *F2I = ws + OFF_F2I, *P2A = ws + OFF_P2A, *P2B = ws + OFF_P2B;
  float *FTR = ws + OFF_FTR, *FTI = ws + OFF_FTI, *COS2 = ws + OFF_COS2, *SIN2 = ws + OFF_SIN2;
  float *XSPEC = ws + OFF_XSPEC, *TVEC = ws + OFF_TVEC, *TCR = ws + OFF_TCR, *TCI = ws + OFF_TCI;
  float *WSP = ws + OFF_WSP, *XS1 = ws + OFF_XS1, *T2R = ws + OFF_T2R, *XS2 = ws + OFF_XS2;
  float *XSP = ws + OFF_XSP, *XCV = ws + OFF_XCV;

  // ---- zero-init regions whose pads/recurrence gaps must be exact zeros
  zerof<<<EWG(SZ_A1)>>>(A1, SZ_A1);
  zerof<<<EWG(SZ_P2)>>>(P2A, SZ_P2);
  zerof<<<EWG(8192)>>>(T1R + 255LL * 8192, 8192);
  zerof<<<EWG(8192)>>>(T1I + 255LL * 8192, 8192);
  zerof<<<EWG(8192)>>>(T2R + 255LL * 8192, 8192);

  // ---- constant tables & conditioning (deterministic per call; graph-safe)
  build_xpad<<<EWG(SZ_XP)>>>(x, XP, SZ_XP);
  k_tvec<<<dim3(1), dim3(256), 0, stream>>>(t_emb, sp_dw, sp_db, drw, drb, diw, dib, TVEC, TCR, TCI);
  k_wc<<<EWG(SZ_WC)>>>(spec_wr, spec_wi, TCR, TCI, WCR, WCI, SZ_WC);
  k_wsp<<<EWG(SZ_WSP)>>>(sp_wt, TVEC, WSP, SZ_WSP);
  k_phase1<<<EWG(SZ_PH1)>>>(COS1, SIN1N, SZ_PH1);
  k_phase2<<<EWG(SZ_PH2)>>>(COS2, SIN2, SZ_PH2);
  k_resize<<<dim3(1), dim3(256), 0, stream>>>(R1, R2P);
  k_legendre<<<dim3(1), dim3(128), 0, stream>>>(A1, 128, 1);     // quadrature-weighted Pw
  k_mirror<<<EWG(127LL * 128 * 128)>>>(A1, 128, 127LL * 128 * 128);
  k_legendre<<<dim3(1), dim3(64), 0, stream>>>(P2A, 64, 0);
  k_mirror<<<EWG(63LL * 64 * 64)>>>(P2A, 64, 63LL * 64 * 64);
  k_transp_p2<<<EWG(SZ_P2)>>>(P2A, P2B, SZ_P2);                  // [m][t][l]

  // ---- SHT forward: g[t][m][c] = phase^T x   (per-t batch, M=256 K=256)
  launch_gemm(stream, COS1,  XP, GR, 256, 256, 256, 128, 128, 0, 256LL * 128, 256LL * 128, 1.0f, 0, 128);
  launch_gemm(stream, SIN1N, XP, GI, 256, 256, 256, 128, 128, 0, 256LL * 128, 256LL * 128, 1.0f, 0, 128);
  // ---- flm1[m][l][c] = Pw[m] @ g[:,m,:]   (per-m batch, strided B)
  launch_gemm(stream, A1, GR, F1R, 128, 128, 128, 256 * 128, 128, 128LL * 128, 128, 128LL * 128, 1.0f, 0, 255);
  launch_gemm(stream, A1, GI, F1I, 128, 128, 128, 256 * 128, 128, 128LL * 128, 128, 128LL * 128, 1.0f, 0, 255);
  // ---- resize dim0 (l: 128 -> 64): T1[m][i][c] = R1 @ F1[m]
  launch_gemm(stream, R1, F1R, T1R, 64, 128, 128, 128, 128, 0, 128LL * 128, 64LL * 128, 1.0f, 0, 255);
  launch_gemm(stream, R1, F1I, T1I, 64, 128, 128, 128, 128, 0, 128LL * 128, 64LL * 128, 1.0f, 0, 255);
  // ---- resize dim1 (m: 255 -> 127): FF[i][j][c] = R2 @ T1[:,i,:]
  launch_gemm(stream, R2P, T1R, FFR, 128, 256, 256, 64 * 128, 128, 0, 128, 128LL * 128, 1.0f, 0, 64);
  launch_gemm(stream, R2P, T1I, FFI, 128, 256, 256, 64 * 128, 128, 0, 128, 128LL * 128, 1.0f, 0, 64);
  // ---- spectral mix (complex, per-l): F2 = FF @ WC
  launch_gemm(stream, FFR, WCR, F2R, 128, 128, 128, 128, 128, 128LL * 128, 128LL * 128, 128LL * 128,  1.0f, 0, 64);
  launch_gemm(stream, FFI, WCI, F2R, 128, 128, 128, 128, 128, 128LL * 128, 128LL * 128, 128LL * 128, -1.0f, 1, 64);
  launch_gemm(stream, FFR, WCI, F2I, 128, 128, 128, 128, 128, 128LL * 128, 128LL * 128, 128LL * 128,  1.0f, 0, 64);
  launch_gemm(stream, FFI, WCR, F2I, 128, 128, 128, 128, 128, 128LL * 128, 128LL * 128, 128LL * 128,  1.0f, 1, 64);
  // ---- inverse SHT step 1: f_t[m][t][c] = P2[m] @ F2[:,m,:]
  launch_gemm(stream, P2B, F2R, FTR, 64, 64, 64, 128 * 128, 128, 64LL * 64, 128, 64LL * 128, 1.0f, 0, 127);
  launch_gemm(stream, P2B, F2I, FTI, 64, 64, 64, 128 * 128, 128, 64LL * 64, 128, 64LL * 128, 1.0f, 0, 127);
  // ---- inverse SHT step 2 (real part): xspec[t] = cos2 @ fr[:,t,:] - sin2 @ fi[:,t,:]
  launch_gemm(stream, COS2, FTR, XSPEC, 128, 128, 128, 64 * 128, 128, 0, 128, 128LL * 128,  1.0f, 0, 64);
  launch_gemm(stream, SIN2, FTI, XSPEC, 128, 128, 128, 64 * 128, 128, 0, 128, 128LL * 128, -1.0f, 1, 64);

  // ---- spatial branch: XS1 = x @ (w_t * sp_weight)
  launch_gemm(stream, x, WSP, XS1, 32640, 128, 128, 128, 128, 0, 0, 0, 1.0f, 0, 1);
  // resize dim0: T2[p][i][c] = R1 @ XS1[:,p,:]
  launch_gemm(stream, R1, XS1, T2R, 64, 128, 128, 255 * 128, 128, 0, 128, 64LL * 128, 1.0f, 0, 255);
  // resize dim1: XS2[i][j][c] = R2 @ T2[:,i,:]
  launch_gemm(stream, R2P, T2R, XS2, 128, 256, 256, 64 * 128, 128, 0, 128, 128LL * 128, 1.0f, 0, 64);
  // zero-pad image, then 3x3 conv (+conv bias + b_t)
  k_xsppad<<<EWG(SZ_XSP)>>>(XS2, XSP, SZ_XSP);
  conv3x3_wmma<<<dim3(2, 8, 64), dim3(32), 0, stream>>>(XSP, sp_ck, sp_cb, TVEC, XCV);

  // ---- fuse: gelu + LayerNorm -> output (64,127,128)
  k_fuse<<<dim3(64 * 127), dim3(128), 0, stream>>>(XSPEC, XCV, lns, lnb, out);
}